// SubpixelConv_15642270892756
// MI455X (gfx1250) — compile-verified
//
#include <hip/hip_runtime.h>

// SubpixelConv: 1x1 conv (M=1048576, K=64, N=256) + BN(inference) + ReLU + depth_to_space(r=2)
// Exact fp32 path via V_WMMA_F32_16X16X4_F32, K chained 16x, dual accumulator chains (nt, nt+4)
// so the two stores per pixel-row land in one contiguous 512B region (rx=0 / rx=1 quadrants).

typedef __attribute__((ext_vector_type(2))) float v2f;
typedef __attribute__((ext_vector_type(8))) float v8f;

#define BN_EPS 1e-3f

constexpr int C_IN   = 64;
constexpr int C_OUT  = 256;
constexpr int M_TOT  = 16 * 256 * 256;   // B*H*W pixels
constexpr int WAVES  = 8;                // 256 threads / wave32
constexpr int BLOCKS = M_TOT / (WAVES * 16);

__global__ __launch_bounds__(256) void subpixel_conv_wmma_f32(
    const float* __restrict__ x,     // [M, 64]
    const float* __restrict__ w,     // [64, 256]
    const float* __restrict__ bias,  // [256]
    const float* __restrict__ gamma, // [256]
    const float* __restrict__ beta,  // [256]
    const float* __restrict__ mean,  // [256]
    const float* __restrict__ var,   // [256]
    float* __restrict__ out)         // [16, 512, 512, 64] flat
{
    // B-matrix fragments pre-swizzled for V_WMMA_F32_16X16X4_F32:
    // fragment (nt, kk): lane&15 = N-in-tile, lane>>4 selects K pair {k0,k0+1}|{k0+2,k0+3}
    // float index = nt*1024 + kk*64 + lane*2 + j   (64 KB total)
    __shared__ float ldsW[16 * 16 * 32 * 2];

    const int tid  = threadIdx.x;
    const int lane = tid & 31;
    const int wave = tid >> 5;

    // Cooperative swizzled fill of w into LDS (w is 64KB, L2-resident across blocks)
    for (int i = tid; i < 16 * 16 * 32; i += 256) {
        const int nt = i >> 9;
        const int kk = (i >> 5) & 15;
        const int l  = i & 31;
        const int k0 = kk * 4 + ((l >> 4) << 1);
        const int n  = nt * 16 + (l & 15);
        ldsW[i * 2 + 0] = w[(k0 + 0) * C_OUT + n];
        ldsW[i * 2 + 1] = w[(k0 + 1) * C_OUT + n];
    }
    __syncthreads();

    const int m_base = (blockIdx.x * WAVES + wave) * 16;

    // A fragments (16x4 f32 layout): row = lane&15, K pair = (lane>>4)*2 + {0,1}
    const int arow = m_base + (lane & 15);
    const int acol = (lane >> 4) << 1;
    const float* xrow = x + (size_t)arow * C_IN + acol;
    v2f afrag[16];
#pragma unroll
    for (int kk = 0; kk < 16; ++kk)
        afrag[kk] = __builtin_nontemporal_load((const v2f*)(xrow + kk * 4));

    // Sweep N-tiles in quadrant pairs (ntA, ntA+4): same ry/co range, rx=0 vs rx=1.
    // Two independent WMMA accumulator chains per iteration for XDL ILP.
    for (int ntp = 0; ntp < 8; ++ntp) {
        const int ntA = (ntp & 3) | ((ntp & 4) << 1);   // 0..3, 8..11
        const float* bptrA = &ldsW[ntA * 1024 + lane * 2];
        const float* bptrB = bptrA + 4 * 1024;          // ntB = ntA + 4

        v8f accA = {0.f, 0.f, 0.f, 0.f, 0.f, 0.f, 0.f, 0.f};
        v8f accB = {0.f, 0.f, 0.f, 0.f, 0.f, 0.f, 0.f, 0.f};
#pragma unroll
        for (int kk = 0; kk < 16; ++kk) {
            v2f bA = *(const v2f*)(bptrA + kk * 64);
            v2f bB = *(const v2f*)(bptrB + kk * 64);
            accA = __builtin_amdgcn_wmma_f32_16x16x4_f32(
                false, afrag[kk], false, bA, (short)0, accA, false, false);
            accB = __builtin_amdgcn_wmma_f32_16x16x4_f32(
                false, afrag[kk], false, bB, (short)0, accB, false, false);
        }

        // Fused bias + BN + ReLU epilogue
        const int oA = ntA * 16 + (lane & 15);
        const int oB = oA + 64;
        const float sA = gamma[oA] * rsqrtf(var[oA] + BN_EPS);
        const float tA = (bias[oA] - mean[oA]) * sA + beta[oA];
        const float sB = gamma[oB] * rsqrtf(var[oB] + BN_EPS);
        const float tB = (bias[oB] - mean[oB]) * sB + beta[oB];
        // oA&64 == 0 by construction; chanB = chanA + 64 (rx quadrant step)
        const unsigned chanA = (((unsigned)(oA & 128)) << 8) + (unsigned)(oA & 63);

#pragma unroll
        for (int r = 0; r < 8; ++r) {
            const int p = m_base + r + ((lane >> 4) << 3);   // pixel index (C/D row layout)
            // pixel p = (b,h,w): base output offset ((b*2H + 2h)*2W + 2w)*64
            const unsigned P = (((unsigned)p & 0xFFFFFF00u) << 8) | (((unsigned)p & 255u) << 7);
            float vA = accA[r] * sA + tA;
            float vB = accB[r] * sB + tB;
            vA = vA > 0.f ? vA : 0.f;
            vB = vB > 0.f ? vB : 0.f;
            __builtin_nontemporal_store(vA, out + (size_t)P + chanA);
            __builtin_nontemporal_store(vB, out + (size_t)P + chanA + 64);
        }
    }
}

extern "C" void kernel_launch(void* const* d_in, const int* in_sizes, int n_in,
                              void* d_out, int out_size, void* d_ws, size_t ws_size,
                              hipStream_t stream) {
    const float* x     = (const float*)d_in[0];
    const float* w     = (const float*)d_in[1];
    const float* b     = (const float*)d_in[2];
    const float* gamma = (const float*)d_in[3];
    const float* beta  = (const float*)d_in[4];
    const float* mmean = (const float*)d_in[5];
    const float* mvar  = (const float*)d_in[6];
    float* out = (float*)d_out;

    hipLaunchKernelGGL(subpixel_conv_wmma_f32,
                       dim3(BLOCKS), dim3(256), 0, stream,
                       x, w, b, gamma, beta, mmean, mvar, out);
}